// SAGEConv_128849019137
// MI455X (gfx1250) — compile-verified
//
#include <hip/hip_runtime.h>

typedef __attribute__((ext_vector_type(2))) float v2f;
typedef __attribute__((ext_vector_type(8))) float v8f;

#define NN 100000      // nodes (== 6250 * 16, no M tail)
#define NE 1600000     // edges
#define D  128         // D_IN == D_OUT

// ---------------------------------------------------------------------------
// Kernel 1: zero the workspace (neigh_sum [NN*128] + deg [NN]) as float4s.
// ---------------------------------------------------------------------------
__global__ __launch_bounds__(256) void k_zero(float4* __restrict__ ws, int n4) {
    int i = blockIdx.x * 256 + threadIdx.x;
    if (i < n4) ws[i] = make_float4(0.f, 0.f, 0.f, 0.f);
}

// ---------------------------------------------------------------------------
// Kernel 2: edge scatter. One wave32 per edge: lane L loads float4
// feat[src][4L..4L+3] (coalesced global_load_b128) and atomically adds it
// into neigh_sum[dst]. Hardware fp32 atomics: entire table is L2-resident.
// ---------------------------------------------------------------------------
__global__ __launch_bounds__(256) void k_scatter(const float* __restrict__ feat,
                                                 const int* __restrict__ src,
                                                 const int* __restrict__ dst,
                                                 float* __restrict__ nsum,
                                                 float* __restrict__ deg) {
    int wave = threadIdx.x >> 5;
    int lane = threadIdx.x & 31;
    int e = blockIdx.x * 8 + wave;
    if (e >= NE) return;
    int s = src[e];
    int d = dst[e];
    float4 v = ((const float4*)(feat + (size_t)s * D))[lane];
    float* o = nsum + (size_t)d * D + lane * 4;
    unsafeAtomicAdd(o + 0, v.x);
    unsafeAtomicAdd(o + 1, v.y);
    unsafeAtomicAdd(o + 2, v.z);
    unsafeAtomicAdd(o + 3, v.w);
    if (lane == 0) unsafeAtomicAdd(deg + d, 1.0f);
}

// ---------------------------------------------------------------------------
// Kernel 3: h_neigh = neigh_sum / max(deg, 1), in place, float4 per thread.
// ---------------------------------------------------------------------------
__global__ __launch_bounds__(256) void k_mean(float* __restrict__ nsum,
                                              const float* __restrict__ deg) {
    int i = blockIdx.x * 256 + threadIdx.x;          // float4 index
    if (i >= NN * (D / 4)) return;
    int row = i >> 5;                                 // / (D/4)
    float inv = 1.0f / fmaxf(deg[row], 1.0f);
    float4* p = (float4*)nsum;
    float4 v = p[i];
    v.x *= inv; v.y *= inv; v.z *= inv; v.w *= inv;
    p[i] = v;
}

// ---------------------------------------------------------------------------
// Kernel 4: rst = [feat | h_neigh] @ W (256x128) + b, native fp32 WMMA.
// grid.x = NN/16 (M tiles); 8 waves per block cover the 8 N-tiles of 16.
// Per wave: 16x16 f32 accumulator (v8f), K-loop of 64 steps of K=4 using
// V_WMMA_F32_16X16X4_F32.
//   A frag (16x4 f32): lanes 0-15 hold M=lane, K={k0,k0+1};
//                      lanes 16-31 hold M=lane-16, K={k0+2,k0+3}  -> one b64 load
//   B frag (4x16 f32): lane group holds B[k][n0+lane&15] for its two K's
//   C/D: VGPR r = D[m0 + (lane>>4)*8 + r][n0 + (lane&15)]
// ---------------------------------------------------------------------------
__global__ __launch_bounds__(256) void k_gemm(const float* __restrict__ feat,
                                              const float* __restrict__ hne,
                                              const float* __restrict__ W,
                                              const float* __restrict__ bias,
                                              float* __restrict__ out) {
    const int tid    = threadIdx.x;
    const int wave   = tid >> 5;
    const int lane   = tid & 31;
    const int laneLo = lane & 15;
    const int laneHi = lane >> 4;          // 0 or 1
    const int m0     = blockIdx.x * 16;
    const int n0     = wave * 16;
    const int row    = m0 + laneLo;

    const float bv = bias[n0 + laneLo];
    v8f c = {bv, bv, bv, bv, bv, bv, bv, bv};

    const float* __restrict__ arow = feat + (size_t)row * D;
    const float* __restrict__ hrow = hne  + (size_t)row * D;
    const float* __restrict__ wcol = W + n0 + laneLo;

    // K = 0..127 : self features
#pragma unroll 8
    for (int k0 = 0; k0 < D; k0 += 4) {
        const int ka = k0 + 2 * laneHi;
        v2f a = *(const v2f*)(arow + ka);
        v2f b;
        b.x = wcol[(size_t)ka * D];
        b.y = wcol[(size_t)(ka + 1) * D];
        c = __builtin_amdgcn_wmma_f32_16x16x4_f32(false, a, false, b,
                                                  (short)0, c, false, false);
    }
    // K = 128..255 : mean-aggregated neighbor features
#pragma unroll 8
    for (int k0 = 0; k0 < D; k0 += 4) {
        const int ka = k0 + 2 * laneHi;
        v2f a = *(const v2f*)(hrow + ka);
        const int kw = D + ka;
        v2f b;
        b.x = wcol[(size_t)kw * D];
        b.y = wcol[(size_t)(kw + 1) * D];
        c = __builtin_amdgcn_wmma_f32_16x16x4_f32(false, a, false, b,
                                                  (short)0, c, false, false);
    }

    float* __restrict__ orow = out + (size_t)(m0 + laneHi * 8) * D + n0 + laneLo;
#pragma unroll
    for (int r = 0; r < 8; ++r)
        orow[(size_t)r * D] = c[r];
}

// ---------------------------------------------------------------------------
extern "C" void kernel_launch(void* const* d_in, const int* in_sizes, int n_in,
                              void* d_out, int out_size, void* d_ws, size_t ws_size,
                              hipStream_t stream) {
    const float* feat = (const float*)d_in[0];
    const int*   src  = (const int*)d_in[1];
    const int*   dst  = (const int*)d_in[2];
    const float* W    = (const float*)d_in[3];
    const float* bias = (const float*)d_in[4];
    float*       out  = (float*)d_out;

    float* nsum = (float*)d_ws;                    // NN * 128 floats
    float* deg  = nsum + (size_t)NN * D;           // NN floats

    const int n4 = NN * (D + 1) / 4;               // zero both regions as float4
    k_zero<<<(n4 + 255) / 256, 256, 0, stream>>>((float4*)d_ws, n4);

    k_scatter<<<(NE + 7) / 8, 256, 0, stream>>>(feat, src, dst, nsum, deg);

    const int nmean = NN * (D / 4);
    k_mean<<<(nmean + 255) / 256, 256, 0, stream>>>(nsum, deg);

    k_gemm<<<NN / 16, 256, 0, stream>>>(feat, nsum, W, bias, out);
}